// REINFORCELoss_72318659330424
// MI455X (gfx1250) — compile-verified
//
#include <hip/hip_runtime.h>
#include <math.h>

#define GAMMA   0.99f
#define T_LEN   2048
#define B_ROWS  4096
#define NCHUNK  16          // T_LEN / 128 timesteps per wave-iteration
#define EPSN    1e-8f

typedef __attribute__((ext_vector_type(4))) float f32x4;

// ---------------------------------------------------------------------------
// TDM compile probe (NOT launched): demonstrates the gfx1250 Tensor Data Mover
// path (tensor_load_to_lds + s_wait_tensorcnt). Deliberately not used in the
// hot path: data here is wave-private, consumed once, and needs no transpose,
// so a memory->LDS->VGPR round trip would only add DS traffic; the register
// pipeline below is the better latency-hiding mechanism for pure streaming.
// ---------------------------------------------------------------------------
#if defined(__has_builtin)
#if __has_builtin(__builtin_amdgcn_tensor_load_to_lds)
typedef __attribute__((ext_vector_type(4))) unsigned int u32x4_t;
typedef __attribute__((ext_vector_type(8))) int          i32x8_t;
typedef __attribute__((ext_vector_type(4))) int          i32x4_t;

__global__ void tdm_probe_kernel() {
  u32x4_t g0 = {0u, 0u, 0u, 0u};
  i32x8_t g1 = {0, 0, 0, 0, 0, 0, 0, 0};
  i32x4_t g2 = {0, 0, 0, 0};
  i32x4_t g3 = {0, 0, 0, 0};
#if __clang_major__ >= 23
  i32x8_t g4 = {0, 0, 0, 0, 0, 0, 0, 0};
  __builtin_amdgcn_tensor_load_to_lds(g0, g1, g2, g3, g4, 0);
#else
  __builtin_amdgcn_tensor_load_to_lds(g0, g1, g2, g3, 0);
#endif
#if __has_builtin(__builtin_amdgcn_s_wait_tensorcnt)
  __builtin_amdgcn_s_wait_tensorcnt(0);
#endif
}
#endif
#endif

// Broadcast lane 0 -> all lanes without an LDS round trip (EXEC is full here,
// so first active lane == lane 0). Lowers to v_readfirstlane_b32.
__device__ __forceinline__ float bcast_lane0(float v) {
  return __int_as_float(__builtin_amdgcn_readfirstlane(__float_as_int(v)));
}

// ---------------------------------------------------------------------------
// Backward affine-scan of one 128-timestep chunk, one wave per row.
// f_t(x) = a_t*x + b_t with a_t = GAMMA*(1-d_t), b_t = r_t; composition
// (a1,b1)o(a2,b2) = (a1*a2, a1*b2 + b1) is associative. C = carry entering
// from the right (R_{chunk_end+1}); updated for the next (lower) chunk.
// ---------------------------------------------------------------------------
__device__ __forceinline__ void wave_scan_chunk(f32x4 r, f32x4 d,
                                                float& C, f32x4& Rout) {
  const int lane = threadIdx.x & 31;

  float a0 = GAMMA - GAMMA * d.x, b0 = r.x;
  float a1 = GAMMA - GAMMA * d.y, b1 = r.y;
  float a2 = GAMMA - GAMMA * d.z, b2 = r.z;
  float a3 = GAMMA - GAMMA * d.w, b3 = r.w;

  // Lane-local composition F = f0 o f1 o f2 o f3 (right-to-left build)
  float Ga = a3, Gb = b3;
  Gb = fmaf(a2, Gb, b2); Ga = a2 * Ga;
  Gb = fmaf(a1, Gb, b1); Ga = a1 * Ga;
  Gb = fmaf(a0, Gb, b0); Ga = a0 * Ga;

  // Backward inclusive scan across the wave; branchless identity-select for
  // out-of-range partners keeps EXEC full (v_cndmask, no exec juggling).
  float Ha = Ga, Hb = Gb;
#pragma unroll
  for (int off = 1; off < 32; off <<= 1) {
    float pa = __shfl_down(Ha, off, 32);
    float pb = __shfl_down(Hb, off, 32);
    const bool ok = (lane + off) < 32;
    pa = ok ? pa : 1.0f;       // identity map (1, 0)
    pb = ok ? pb : 0.0f;
    Hb = fmaf(Ha, pb, Hb);     // H = H_cur o H_right (uses old Ha)
    Ha = Ha * pa;
  }

  // Exclusive-from-right: value entering this lane = H_{lane+1}(C)
  float Ea = __shfl_down(Ha, 1, 32);
  float Eb = __shfl_down(Hb, 1, 32);
  if (lane == 31) { Ea = 1.0f; Eb = 0.0f; }
  float x = fmaf(Ea, C, Eb);

  // Unroll the 4 in-lane returns right-to-left
  float R3 = fmaf(a3, x,  b3);
  float R2 = fmaf(a2, R3, b2);
  float R1 = fmaf(a1, R2, b1);
  float R0 = fmaf(a0, R1, b0);

  // New carry = (inclusive composite at lane 0)(old C) -- SALU broadcast
  float Aa = bcast_lane0(Ha);
  float Ab = bcast_lane0(Hb);
  C = fmaf(Aa, C, Ab);

  f32x4 R; R.x = R0; R.y = R1; R.z = R2; R.w = R3;
  Rout = R;
}

// ---------------------------------------------------------------------------
__global__ void init_acc_kernel(double* __restrict__ acc) {
  acc[0] = 0.0;  // sum R
  acc[1] = 0.0;  // sum R^2
}

// Pass 1: backward scan per row, accumulate global sum / sumsq.
__global__ void __launch_bounds__(256)
reinforce_pass1_kernel(const float* __restrict__ rewards,
                       const float* __restrict__ dones,
                       double* __restrict__ acc) {
  const int gtid = blockIdx.x * blockDim.x + threadIdx.x;
  const int row  = gtid >> 5;
  const int lane = threadIdx.x & 31;

  const f32x4* R4 = (const f32x4*)(rewards + (size_t)row * T_LEN);
  const f32x4* D4 = (const f32x4*)(dones   + (size_t)row * T_LEN);

  // 4-deep register pipeline (slot = c & 3)
  f32x4 rb[4], db[4];
#pragma unroll
  for (int c = NCHUNK - 1; c >= NCHUNK - 4; --c) {
    rb[c & 3] = R4[c * 32 + lane];
    db[c & 3] = D4[c * 32 + lane];
  }

  float C = 0.0f, s1 = 0.0f, s2 = 0.0f;
#pragma unroll
  for (int c = NCHUNK - 1; c >= 0; --c) {
    f32x4 r = rb[c & 3];
    f32x4 d = db[c & 3];
    if (c >= 4) {
      rb[(c - 4) & 3] = R4[(c - 4) * 32 + lane];
      db[(c - 4) & 3] = D4[(c - 4) * 32 + lane];
    }
    if (c >= 8) {   // far prefetch -> global_prefetch_b8
      __builtin_prefetch(&R4[(c - 8) * 32 + lane], 0, 3);
      __builtin_prefetch(&D4[(c - 8) * 32 + lane], 0, 3);
    }
    f32x4 Rv;
    wave_scan_chunk(r, d, C, Rv);
    s1 += Rv.x + Rv.y + Rv.z + Rv.w;
    s2 = fmaf(Rv.x, Rv.x, s2);
    s2 = fmaf(Rv.y, Rv.y, s2);
    s2 = fmaf(Rv.z, Rv.z, s2);
    s2 = fmaf(Rv.w, Rv.w, s2);
  }

  // wave32 reduction, one f64 atomic pair per wave
#pragma unroll
  for (int off = 16; off >= 1; off >>= 1) {
    s1 += __shfl_xor(s1, off, 32);
    s2 += __shfl_xor(s2, off, 32);
  }
  if (lane == 0) {
    atomicAdd(&acc[0], (double)s1);
    atomicAdd(&acc[1], (double)s2);
  }
}

// Pass 2: recompute scan (rewards/done L2-resident after pass 1), normalize,
// multiply with -log_probs. log_probs loads and out stores are non-temporal
// (TH=NT) so single-use traffic does not evict the re-used arrays from L2.
__global__ void __launch_bounds__(256)
reinforce_pass2_kernel(const float* __restrict__ log_probs,
                       const float* __restrict__ rewards,
                       const float* __restrict__ dones,
                       const double* __restrict__ acc,
                       float* __restrict__ out) {
  const double N   = (double)B_ROWS * (double)T_LEN;
  const float mean = (float)(acc[0] / N);
  float var = (float)(acc[1] / N) - mean * mean;
  var = var < 0.0f ? 0.0f : var;
  const float rstd = 1.0f / (sqrtf(var) + EPSN);

  const int gtid = blockIdx.x * blockDim.x + threadIdx.x;
  const int row  = gtid >> 5;
  const int lane = threadIdx.x & 31;

  const f32x4* R4 = (const f32x4*)(rewards   + (size_t)row * T_LEN);
  const f32x4* D4 = (const f32x4*)(dones     + (size_t)row * T_LEN);
  const f32x4* L4 = (const f32x4*)(log_probs + (size_t)row * T_LEN);
  f32x4*       O4 = (f32x4*)(out + (size_t)row * T_LEN);

  f32x4 rb[4], db[4], lb[4];
#pragma unroll
  for (int c = NCHUNK - 1; c >= NCHUNK - 4; --c) {
    rb[c & 3] = R4[c * 32 + lane];
    db[c & 3] = D4[c * 32 + lane];
    lb[c & 3] = __builtin_nontemporal_load(&L4[c * 32 + lane]);
  }

  float C = 0.0f;
#pragma unroll
  for (int c = NCHUNK - 1; c >= 0; --c) {
    f32x4 r  = rb[c & 3];
    f32x4 d  = db[c & 3];
    f32x4 lp = lb[c & 3];
    if (c >= 4) {
      rb[(c - 4) & 3] = R4[(c - 4) * 32 + lane];
      db[(c - 4) & 3] = D4[(c - 4) * 32 + lane];
      lb[(c - 4) & 3] = __builtin_nontemporal_load(&L4[(c - 4) * 32 + lane]);
    }
    if (c >= 8) {
      __builtin_prefetch(&R4[(c - 8) * 32 + lane], 0, 3);
      __builtin_prefetch(&D4[(c - 8) * 32 + lane], 0, 3);
      __builtin_prefetch(&L4[(c - 8) * 32 + lane], 0, 3);
    }
    f32x4 Rv;
    wave_scan_chunk(r, d, C, Rv);

    f32x4 o;
    o.x = -lp.x * ((Rv.x - mean) * rstd);
    o.y = -lp.y * ((Rv.y - mean) * rstd);
    o.z = -lp.z * ((Rv.z - mean) * rstd);
    o.w = -lp.w * ((Rv.w - mean) * rstd);
    __builtin_nontemporal_store(o, &O4[c * 32 + lane]);
  }
}

// ---------------------------------------------------------------------------
extern "C" void kernel_launch(void* const* d_in, const int* in_sizes, int n_in,
                              void* d_out, int out_size, void* d_ws, size_t ws_size,
                              hipStream_t stream) {
  (void)in_sizes; (void)n_in; (void)out_size; (void)ws_size;

  const float* log_probs = (const float*)d_in[0];
  const float* rewards   = (const float*)d_in[1];
  const float* dones     = (const float*)d_in[2];
  double*      acc       = (double*)d_ws;
  float*       out       = (float*)d_out;

  const int threads = 256;                       // 8 waves per workgroup
  const int blocks  = (B_ROWS * 32) / threads;   // one wave per row -> 512

  init_acc_kernel<<<1, 1, 0, stream>>>(acc);
  reinforce_pass1_kernel<<<blocks, threads, 0, stream>>>(rewards, dones, acc);
  reinforce_pass2_kernel<<<blocks, threads, 0, stream>>>(log_probs, rewards,
                                                         dones, acc, out);
}